// GCNLayer_61838939128218
// MI455X (gfx1250) — compile-verified
//
#include <hip/hip_runtime.h>

// GCN layer: h = x@W + b  ->  scatter-mean over edges  ->  BN(eval) + ReLU
// MI455X strategy:
//  - GEMM via V_WMMA_F32_16X16X4_F32 (full fp32 precision; GEMM is only
//    3.28 GFLOP -> negligible vs the edge scatter).
//  - Scatter-mean via fp32 L2 atomics: h (51.2MB) + accumulator (51.2MB,
//    aliased onto d_out) + counts all fit in the 192MB global L2, so the
//    random gather + 204.8M atomic adds stay L2-resident.

#define N_NODES 100000
#define N_EDGES 1600000
#define C       128
#define BN_EPS  1e-5f

typedef __attribute__((ext_vector_type(2))) float v2f;
typedef __attribute__((ext_vector_type(8))) float v8f;

// ---------------------------------------------------------------- zero ----
__global__ void zero_kernel(float* __restrict__ out, float* __restrict__ cnt) {
    int idx = blockIdx.x * blockDim.x + threadIdx.x;
    if (idx < N_NODES * C) out[idx] = 0.0f;
    if (idx < N_NODES)     cnt[idx] = 0.0f;
}

// ---------------------------------------------------------------- gemm ----
// Block = 256 threads = 8 waves. Block computes rows [16*bid, 16*bid+16).
// Wave w computes the 16x16 output tile at columns [16w, 16w+16).
// K loop: 128 in steps of 4 -> 32 x v_wmma_f32_16x16x4_f32, fully unrolled.
//
// ISA 7.12.2 layouts (wave32):
//  A (16x4 f32, 2 VGPRs): lanes 0-15 hold M=lane, K={0,1}; lanes 16-31 hold
//    M=lane-16, K={2,3}.
//  B (4x16 f32, 2 VGPRs): row-per-half-wave mirror: lanes 0-15 N=lane with
//    K={0,1}; lanes 16-31 N=lane-16 with K={2,3}.
//  C/D (16x16 f32, 8 VGPRs): VGPR v: lanes 0-15 -> M=v, lanes 16-31 -> M=v+8,
//    N = lane%16.
__global__ __launch_bounds__(256) void gemm_kernel(
    const float* __restrict__ x, const float* __restrict__ W,
    const float* __restrict__ bias, float* __restrict__ h) {

    const int lane  = threadIdx.x & 31;
    const int wave  = threadIdx.x >> 5;          // 0..7 => N tile
    const int row0  = blockIdx.x * 16;
    const int ncol  = wave * 16;
    const int m     = lane & 15;                 // row within tile (A & C)
    const int n     = lane & 15;                 // col within tile (B & C)
    const int khalf = (lane >> 4) << 1;          // 0 (lanes 0-15) or 2

    const float* __restrict__ xrow = x + (size_t)(row0 + m) * C;

    v8f acc = {};
#pragma unroll
    for (int kk = 0; kk < C; kk += 4) {
        v2f a, b;
        a.x = xrow[kk + khalf];
        a.y = xrow[kk + khalf + 1];
        b.x = W[(size_t)(kk + khalf)     * C + ncol + n];
        b.y = W[(size_t)(kk + khalf + 1) * C + ncol + n];
        // 8 args: (neg_a, A, neg_b, B, c_mod, C, reuse_a, reuse_b)
        acc = __builtin_amdgcn_wmma_f32_16x16x4_f32(
            false, a, false, b, (short)0, acc, false, false);
    }

    const int   mhi  = (lane >> 4) * 8;          // 0 or 8
    const float bval = bias[ncol + n];
#pragma unroll
    for (int v = 0; v < 8; ++v) {
        h[(size_t)(row0 + v + mhi) * C + ncol + n] = acc[v] + bval;
    }
}

// ------------------------------------------------------------- scatter ----
// One wave per edge: 32 lanes x 4 channels = 128 channels.
// float4 gather from h[src], 4 relaxed agent-scope fp32 atomic adds into
// out[dst] (non-returning -> global_atomic_add_f32, STOREcnt only).
__global__ __launch_bounds__(256) void scatter_kernel(
    const int* __restrict__ edges, const float* __restrict__ h,
    float* __restrict__ out, float* __restrict__ cnt) {

    const int wid  = (blockIdx.x * blockDim.x + threadIdx.x) >> 5;
    const int lane = threadIdx.x & 31;
    if (wid >= N_EDGES) return;

    const int src = edges[wid];               // edge_index[0][e]
    const int dst = edges[N_EDGES + wid];     // edge_index[1][e]

    const float4 v = *(const float4*)(h + (size_t)src * C + lane * 4);
    float* o = out + (size_t)dst * C + lane * 4;
    __hip_atomic_fetch_add(o + 0, v.x, __ATOMIC_RELAXED, __HIP_MEMORY_SCOPE_AGENT);
    __hip_atomic_fetch_add(o + 1, v.y, __ATOMIC_RELAXED, __HIP_MEMORY_SCOPE_AGENT);
    __hip_atomic_fetch_add(o + 2, v.z, __ATOMIC_RELAXED, __HIP_MEMORY_SCOPE_AGENT);
    __hip_atomic_fetch_add(o + 3, v.w, __ATOMIC_RELAXED, __HIP_MEMORY_SCOPE_AGENT);
    if (lane == 0)
        __hip_atomic_fetch_add(cnt + dst, 1.0f, __ATOMIC_RELAXED, __HIP_MEMORY_SCOPE_AGENT);
}

// ------------------------------------------------------------ finalize ----
__global__ void finalize_kernel(
    float* __restrict__ out, const float* __restrict__ cnt,
    const float* __restrict__ gamma, const float* __restrict__ beta,
    const float* __restrict__ rm, const float* __restrict__ rv) {

    int idx = blockIdx.x * blockDim.x + threadIdx.x;
    if (idx >= N_NODES * C) return;
    int c    = idx & (C - 1);
    int node = idx >> 7;

    float agg = out[idx] / fmaxf(cnt[node], 1.0f);
    float inv = rsqrtf(rv[c] + BN_EPS);
    float val = (agg - rm[c]) * (inv * gamma[c]) + beta[c];
    out[idx]  = fmaxf(val, 0.0f);
}

// -------------------------------------------------------------- launch ----
extern "C" void kernel_launch(void* const* d_in, const int* in_sizes, int n_in,
                              void* d_out, int out_size, void* d_ws, size_t ws_size,
                              hipStream_t stream) {
    const float* x     = (const float*)d_in[0];
    const int*   edges = (const int*)  d_in[1];
    const float* W     = (const float*)d_in[2];
    const float* b     = (const float*)d_in[3];
    const float* gamma = (const float*)d_in[4];
    const float* beta  = (const float*)d_in[5];
    const float* rm    = (const float*)d_in[6];
    const float* rv    = (const float*)d_in[7];

    float* out = (float*)d_out;                       // doubles as sum buffer
    float* h   = (float*)d_ws;                        // [N_NODES, C] fp32
    float* cnt = h + (size_t)N_NODES * C;             // [N_NODES]

    zero_kernel<<<(N_NODES * C + 255) / 256, 256, 0, stream>>>(out, cnt);

    gemm_kernel<<<N_NODES / 16, 256, 0, stream>>>(x, W, b, h);

    const long long sthreads = (long long)N_EDGES * 32;
    scatter_kernel<<<(unsigned)((sthreads + 255) / 256), 256, 0, stream>>>(
        edges, h, out, cnt);

    finalize_kernel<<<(N_NODES * C + 255) / 256, 256, 0, stream>>>(
        out, cnt, gamma, beta, rm, rv);
}